// MyModel_61933428410031
// MI455X (gfx1250) — compile-verified
//
#include <hip/hip_runtime.h>
#include <hip/hip_bf16.h>
#include <stdint.h>

#define TILE_ROWS 8
#define NCLASS    1024

typedef unsigned int u32;
typedef unsigned long long u64;
typedef __attribute__((ext_vector_type(4))) u32 su32x4;
typedef __attribute__((ext_vector_type(8))) u32 su32x8;

// ---------------------------------------------------------------------------
// Issue a TDM load of an (nrows x 1024) fp32 tile (row-contiguous) into LDS.
// D# layout per CDNA5 ISA ch.8 (group0: 4 SGPRs, group1: 8 SGPRs, 2D tensor
// so VADDR2/VADDR3 = NULL). Tracked with TENSORcnt.
// ---------------------------------------------------------------------------
__device__ __forceinline__ void tdm_load_tile(const float* gsrc, u32 lds_byte_off, u32 nrows)
{
    union { su32x4 v; u32 u[4]; } g0;
    union { su32x8 v; u32 u[8]; } g1;
    unsigned long long ga = (unsigned long long)(uintptr_t)gsrc;

    g0.u[0] = 1u;                                     // count=1 (valid user D#)
    g0.u[1] = lds_byte_off;                           // lds_addr
    g0.u[2] = (u32)(ga & 0xFFFFFFFFull);              // global_addr[31:0]
    g0.u[3] = (u32)((ga >> 32) & 0x01FFFFFFull)       // global_addr[56:32]
            | (2u << 30);                             // type = 2 ("image")

    g1.u[0] = (2u << 16);                             // data_size=2 -> 4 bytes
    g1.u[1] = ((u32)NCLASS & 0xFFFFu) << 16;          // tensor_dim0[15:0]
    g1.u[2] = ((u32)NCLASS >> 16)                     // tensor_dim0[31:16]
            | ((nrows & 0xFFFFu) << 16);              // tensor_dim1[15:0]
    g1.u[3] = ((u32)NCLASS << 16);                    // tensor_dim1[31:16]=0 | tile_dim0=1024
    g1.u[4] = (nrows & 0xFFFFu);                      // tile_dim1 | tile_dim2=0
    g1.u[5] = (u32)NCLASS;                            // tensor_dim0_stride[31:0]
    g1.u[6] = 0u;                                     // stride hi / dim1_stride lo
    g1.u[7] = 0u;                                     // dim1_stride hi

    asm volatile("tensor_load_to_lds %0, %1" :: "s"(g0.v), "s"(g1.v) : "memory");
}

// ---------------------------------------------------------------------------
// Main pass: one wave per row, TDM double-buffered tiles in LDS.
// Accumulates per-class and total per-sample NLL in 2^-32 fixed point (u64
// atomics -> order-independent, deterministic across replays).
// ws layout: u64 cls[1024], u64 total
// ---------------------------------------------------------------------------
extern "C" __global__ __launch_bounds__(256, 1)
void nll_rows_kernel(const float* __restrict__ x,
                     const long long* __restrict__ tgt,
                     u64* __restrict__ cls_sum,
                     u64* __restrict__ tot_sum,
                     int numTiles)
{
    __shared__ float buf[2][TILE_ROWS][NCLASS];       // exactly 64 KB
    const int tid  = threadIdx.x;
    const int wid  = tid >> 5;
    const int lane = tid & 31;
    const u32 lds0 = (u32)(uintptr_t)(&buf[0][0][0]); // low 32 bits = LDS offset

    const long long tile0 = (long long)blockIdx.x;
    if (wid == 0 && tile0 < numTiles)
        tdm_load_tile(x + tile0 * (long long)(TILE_ROWS * NCLASS), lds0, TILE_ROWS);

    u64 acc = 0ull;
    int it = 0;
    for (long long tile = tile0; tile < numTiles; tile += gridDim.x, ++it) {
        if (wid == 0) {
            long long nxt = tile + gridDim.x;
            if (nxt < numTiles) {
                tdm_load_tile(x + nxt * (long long)(TILE_ROWS * NCLASS),
                              lds0 + (u32)((it + 1) & 1) * (TILE_ROWS * NCLASS * 4u),
                              TILE_ROWS);
                __builtin_amdgcn_s_wait_tensorcnt(1);  // current tile resident
            } else {
                __builtin_amdgcn_s_wait_tensorcnt(0);
            }
        }
        __syncthreads();

        const float*  row  = &buf[it & 1][wid][0];
        const float4* row4 = (const float4*)row;
        float4 v[8];
        #pragma unroll
        for (int i = 0; i < 8; ++i) v[i] = row4[i * 32 + lane];   // ds_load_b128

        // row max
        float m = -3.402823466e+38f;
        #pragma unroll
        for (int i = 0; i < 8; ++i)
            m = fmaxf(m, fmaxf(fmaxf(v[i].x, v[i].y), fmaxf(v[i].z, v[i].w)));
        #pragma unroll
        for (int off = 16; off >= 1; off >>= 1)
            m = fmaxf(m, __shfl_xor(m, off, 32));

        // sum exp(x - max)
        float s = 0.0f;
        #pragma unroll
        for (int i = 0; i < 8; ++i) {
            s += __expf(v[i].x - m); s += __expf(v[i].y - m);
            s += __expf(v[i].z - m); s += __expf(v[i].w - m);
        }
        #pragma unroll
        for (int off = 16; off >= 1; off >>= 1)
            s += __shfl_xor(s, off, 32);

        if (lane == 0) {
            long long r = tile * TILE_ROWS + wid;
            int   t   = (int)tgt[r];
            float lse = m + __logf(s);
            float ps  = lse - row[t];                  // per-sample NLL, >= 0
            u64   q   = (u64)((double)ps * 4294967296.0);  // 2^32 fixed point
            atomicAdd(&cls_sum[t], q);
            acc += q;
        }
        __syncthreads();                               // buffer free for next TDM
    }

    // block reduction of per-wave totals (reuse tile LDS)
    u64* part = (u64*)&buf[0][0][0];
    if (lane == 0) part[wid] = acc;
    __syncthreads();
    if (tid == 0) {
        u64 s = 0ull;
        #pragma unroll
        for (int w = 0; w < TILE_ROWS; ++w) s += part[w];
        atomicAdd(tot_sum, s);
    }
}

extern "C" __global__ void nll_init_kernel(u64* ws)
{
    int i = blockIdx.x * blockDim.x + threadIdx.x;
    if (i < NCLASS + 1) ws[i] = 0ull;
}

// incorrect = (sum over classes of f32(per_class)) / N ; correct = f32(total)/N
extern "C" __global__ void nll_final_kernel(const u64* __restrict__ ws,
                                            float* __restrict__ out, float invN)
{
    if (threadIdx.x == 0 && blockIdx.x == 0) {
        const double inv32 = 1.0 / 4294967296.0;
        float s = 0.0f;
        for (int c = 0; c < NCLASS; ++c)
            s += (float)((double)ws[c] * inv32);       // per-class f32, f32 sum
        float incorrect = s * invN;
        float correct   = (float)((double)ws[NCLASS] * inv32) * invN;
        out[0] = fabsf(incorrect - correct);
    }
}

extern "C" void kernel_launch(void* const* d_in, const int* in_sizes, int n_in,
                              void* d_out, int out_size, void* d_ws, size_t ws_size,
                              hipStream_t stream)
{
    const float*     x   = (const float*)d_in[0];
    const long long* tgt = (const long long*)d_in[1];
    u64*   ws  = (u64*)d_ws;        // [0..1023] class sums, [1024] total
    float* out = (float*)d_out;

    const int N        = in_sizes[1];
    const int numTiles = N / TILE_ROWS;
    int grid = 2048;
    if (grid > numTiles) grid = (numTiles > 0) ? numTiles : 1;

    nll_init_kernel<<<dim3((NCLASS + 1 + 255) / 256), dim3(256), 0, stream>>>(ws);
    nll_rows_kernel<<<dim3(grid), dim3(256), 0, stream>>>(x, tgt, ws, ws + NCLASS, numTiles);
    nll_final_kernel<<<dim3(1), dim3(64), 0, stream>>>(ws, out, 1.0f / (float)N);
}